// Dynamic_MultiTeacher5_52699248721950
// MI455X (gfx1250) — compile-verified
//
#include <hip/hip_runtime.h>
#include <math.h>

// ---------------------------------------------------------------------------
// Dynamic multi-teacher KD loss for MI455X (gfx1250, wave32).
//
// Roofline: 6 x 8192 x 1000 f32 = ~197 MB mandatory reads -> ~8.4 us at
// 23.3 TB/s. Bandwidth bound, so every input matrix is read from HBM exactly
// once: 16 rows/block, out_s tile + current-teacher tile + running mimic-sum
// tile staged in LDS (3 * 16 * 1004 * 4B ~= 193 KB < 320 KB/WGP).
//
// The per-row reduction  S[b] = sum_c exp(t[b,c]/20) * s[b,c]  is done on the
// matrix pipe: V_WMMA_F32_16X16X4_F32 with A = exp(teacher/20) tile and
// B = out_s tile (k-sliced); the accumulator diagonal D[m][m] is exactly S.
// Z[b] = sum_c exp(t[b,c]/20) rides along as per-lane adds.
// exp(x/20) needs no max-subtraction (inputs ~N(0,1)), so one pass suffices.
//
// Polish vs round 0: wave index is forced into an SGPR (readfirstlane) so the
// chunk loop runs on SALU control flow with uniform LDS bases, and the WMMA
// loop has a compile-time trip count (15 + 10-wave tail) so it can unroll.
// ---------------------------------------------------------------------------

#define BB      8192
#define CC      1000
#define RS      1004        // padded LDS row stride in floats (bank-conflict free, 16B aligned)
#define ROWS    16          // rows per block (= WMMA M/N)
#define NCHUNK  250         // CC / 4 (WMMA K = 4)
#define BLK1    512         // 16 waves of 32
#define NW      16

typedef __attribute__((ext_vector_type(2))) float v2f;
typedef __attribute__((ext_vector_type(8))) float v8f;

static_assert(CC % 4 == 0, "C must be divisible by WMMA K");
static_assert(NCHUNK == 15 * NW + 10, "tail split assumption");

__device__ __forceinline__ void top2_merge(float& m1, float& m2, float om1, float om2) {
    if (om1 > m1) { m2 = fmaxf(m1, om2); m1 = om1; }
    else          { m2 = fmaxf(m2, om1); }
}

__global__ __launch_bounds__(BLK1)
void kd_row_kernel(const float* __restrict__ g0, const float* __restrict__ g1,
                   const float* __restrict__ g2, const float* __restrict__ g3,
                   const float* __restrict__ g4, const float* __restrict__ gs,
                   const int*   __restrict__ targets,
                   float* __restrict__ ws) {
    extern __shared__ float smem[];
    float* s_s = smem;                 // out_s tile      [16][RS]
    float* s_t = smem + ROWS * RS;     // teacher tile    [16][RS]
    float* s_m = smem + 2 * ROWS * RS; // mimic-sum tile  [16][RS]

    __shared__ float s_margin[6][ROWS];
    __shared__ float s_tgtl[6][ROWS];
    __shared__ float s_kd[6][ROWS];
    __shared__ float s_ce[ROWS];
    __shared__ float s_lse20[ROWS];
    __shared__ float s_tm1[5][ROWS];       // per-teacher row max (for global max)
    __shared__ float s_sred[NW][ROWS];     // per-wave WMMA diagonal partials
    __shared__ float s_zred[NW][ROWS];     // per-wave Z partials
    __shared__ int   s_tgtidx[ROWS];

    const int tid  = threadIdx.x;
    const int lane = tid & 31;
    // wave index is uniform across the wave: pin it to an SGPR so loops and
    // LDS row bases become scalar (SALU loop control, uniform DS addressing).
    const int wave = __builtin_amdgcn_readfirstlane(tid >> 5);
    const int row0 = blockIdx.x * ROWS;

    // ---- stage out_s tile (coalesced float4) + target indices ----
    for (int i = tid; i < ROWS * NCHUNK; i += BLK1) {
        int row = i / NCHUNK, c4 = (i % NCHUNK) * 4;
        *(float4*)(s_s + row * RS + c4) =
            *(const float4*)(gs + (size_t)(row0 + row) * CC + c4);
    }
    if (tid < ROWS) s_tgtidx[tid] = targets[row0 + tid];
    __syncthreads();

    // ---- out_s row stats: CE pieces and LSE at T=20 (wave w <-> row w) ----
    {
        const float* sp = s_s + wave * RS;
        float mx = -INFINITY;
        for (int c = lane; c < CC; c += 32) mx = fmaxf(mx, sp[c]);
        for (int off = 16; off; off >>= 1) mx = fmaxf(mx, __shfl_xor(mx, off, 32));
        float e1 = 0.f, e20 = 0.f;
        for (int c = lane; c < CC; c += 32) {
            float v = sp[c];
            e1  += __expf(v - mx);
            e20 += __expf((v - mx) * 0.05f);
        }
        for (int off = 16; off; off >>= 1) {
            e1  += __shfl_xor(e1,  off, 32);
            e20 += __shfl_xor(e20, off, 32);
        }
        if (lane == 0) {
            float st = sp[s_tgtidx[wave]];
            s_ce[wave]    = (mx + __logf(e1)) - st;        // CE = LSE1(s) - s[target]
            s_lse20[wave] = mx * 0.05f + __logf(e20);      // log sum exp(s/20)
        }
    }

    // ---- 6 sources: teachers 0..4 then mimic (sum/5 from LDS accumulator) ----
    for (int src = 0; src < 6; ++src) {
        __syncthreads();   // protect tile restage + sred/zred reuse

        const float* tp;
        float vscale, escale;
        if (src < 5) {
            const float* gt = src == 0 ? g0 : src == 1 ? g1 : src == 2 ? g2
                            : src == 3 ? g3 : g4;
            const float* gn = src == 0 ? g1 : src == 1 ? g2 : src == 2 ? g3 : g4;
            for (int i = tid; i < ROWS * NCHUNK; i += BLK1) {
                int row = i / NCHUNK, c4 = (i % NCHUNK) * 4;
                size_t goff = (size_t)(row0 + row) * CC + c4;
                float4 v = *(const float4*)(gt + goff);
                *(float4*)(s_t + row * RS + c4) = v;
                float4* dm = (float4*)(s_m + row * RS + c4);
                if (src == 0) { *dm = v; }
                else { float4 m = *dm; m.x += v.x; m.y += v.y; m.z += v.z; m.w += v.w; *dm = m; }
                if (src < 4) __builtin_prefetch(gn + goff, 0, 1);  // global_prefetch_b8
            }
            __syncthreads();
            tp = s_t; vscale = 1.0f;  escale = 0.05f;   // exp(t/20)
        } else {
            tp = s_m; vscale = 0.2f;  escale = 0.01f;   // exp((msum/5)/20)
        }

        // -- margin / target-logit stats (wave w <-> row w) --
        {
            const float* rp = tp + wave * RS;
            float m1 = -INFINITY, m2 = -INFINITY;
            for (int c = lane; c < CC; c += 32) {
                float v = vscale * rp[c];
                if (v > m1) { m2 = m1; m1 = v; } else if (v > m2) { m2 = v; }
            }
            for (int off = 16; off; off >>= 1) {
                float om1 = __shfl_xor(m1, off, 32);
                float om2 = __shfl_xor(m2, off, 32);
                top2_merge(m1, m2, om1, om2);
            }
            if (lane == 0) {
                float tv = vscale * rp[s_tgtidx[wave]];
                s_margin[src][wave] = (tv == m1) ? (m1 - m2) : 0.0f;
                s_tgtl[src][wave]   = tv;
                if (src < 5) s_tm1[src][wave] = m1;
            }
        }

        // -- WMMA pass: acc diag = S[row] = sum_c exp(t/20)*s ; Z via lane adds --
        {
            const int arow = lane & 15;          // A row (M) and B col (N) for this lane
            const int koff = (lane >> 4) << 1;   // k offset within chunk: 0 or 2
            const float* tb = tp  + arow * RS + koff;
            const float* sb = s_s + arow * RS + koff;
            float zacc = 0.f;
            v8f acc = {};
            // chunks j = wave + 16*i : constant trip count -> unrollable
#pragma unroll 5
            for (int i = 0; i < 15; ++i) {
                int cb = 4 * (wave + (i << 4));
                v2f a, b;
                a.x = __expf(escale * tb[cb]);
                a.y = __expf(escale * tb[cb + 1]);
                b.x = sb[cb];
                b.y = sb[cb + 1];
                zacc += a.x + a.y;
                // D(16x16,f32) += A(16x4,f32) x B(4x16,f32)
                acc = __builtin_amdgcn_wmma_f32_16x16x4_f32(
                          false, a, false, b, (short)0, acc, false, false);
            }
            if (wave < 10) {                     // tail chunks 240..249 (wave-uniform branch)
                int cb = 4 * (wave + 240);
                v2f a, b;
                a.x = __expf(escale * tb[cb]);
                a.y = __expf(escale * tb[cb + 1]);
                b.x = sb[cb];
                b.y = sb[cb + 1];
                zacc += a.x + a.y;
                acc = __builtin_amdgcn_wmma_f32_16x16x4_f32(
                          false, a, false, b, (short)0, acc, false, false);
            }
            float zrow = zacc + __shfl_xor(zacc, 16, 32);
            if (lane < 16) s_zred[wave][lane] = zrow;

            // diagonal extraction per 16x16 f32 C/D layout:
            // D[m][m]: m<8 -> VGPR m, lane m ; m>=8 -> VGPR m-8, lane m+16
            float dval = 0.f;
#pragma unroll
            for (int r = 0; r < 8; ++r) {
                float v = acc[r];
                if (lane == r)      dval = v;   // row r
                if (lane == 24 + r) dval = v;   // row r+8 (N = lane-16 = r+8)
            }
            if (lane < 8)        s_sred[wave][lane]      = dval;
            else if (lane >= 24) s_sred[wave][lane - 16] = dval;
        }
        __syncthreads();

        // -- combine wave partials -> KD[src][row] --
        if (tid < ROWS) {
            float S = 0.f, Z = 0.f;
            for (int w = 0; w < NW; ++w) { S += s_sred[w][tid]; Z += s_zred[w][tid]; }
            // KD = -T^2 * sum_c p_t * log_p_sT = -20*(S/Z) + 400*LSE20(s)
            s_kd[src][tid] = -20.0f * (S / Z) + 400.0f * s_lse20[tid];
        }
    }
    __syncthreads();

    // ---- write per-row results: [margins x6 | tgt x6 | kd x6 | CE | pad] ----
    if (tid < ROWS) {
        float* o = ws + (size_t)(row0 + tid) * 20;
#pragma unroll
        for (int k = 0; k < 6; ++k) {
            o[k]      = s_margin[k][tid];
            o[6 + k]  = s_tgtl[k][tid];
            o[12 + k] = s_kd[k][tid];
        }
        o[18] = s_ce[tid];
    }
    if (tid == 0) {
        float g = -INFINITY;
        for (int k = 0; k < 5; ++k)
            for (int r = 0; r < ROWS; ++r) g = fmaxf(g, s_tm1[k][r]);
        ws[(size_t)BB * 20 + blockIdx.x] = g;   // per-block teacher max
    }
}

__global__ __launch_bounds__(256)
void kd_final_kernel(const float* __restrict__ ws, float* __restrict__ out) {
    __shared__ float  smax[8];
    __shared__ double ssum[8];
    const int tid = threadIdx.x, lane = tid & 31, wv = tid >> 5;
    const int nblk = BB / ROWS;

    // global teacher max over per-block maxes
    float mx = -INFINITY;
    for (int i = tid; i < nblk; i += 256) mx = fmaxf(mx, ws[(size_t)BB * 20 + i]);
    for (int off = 16; off; off >>= 1) mx = fmaxf(mx, __shfl_xor(mx, off, 32));
    if (lane == 0) smax[wv] = mx;
    __syncthreads();
    float maxp = -INFINITY;
    for (int w = 0; w < 8; ++w) maxp = fmaxf(maxp, smax[w]);

    double acc = 0.0;
    for (int r = tid; r < BB; r += 256) {
        const float* o = ws + (size_t)r * 20;
        float mg[6], tg[6], kd[6];
#pragma unroll
        for (int k = 0; k < 6; ++k) { mg[k] = o[k]; tg[k] = o[6 + k]; kd[k] = o[12 + k]; }
        float ce = o[18];
        // softmax over sources of margins / T_TH (T_TH = 2)
        float xmax = -INFINITY;
#pragma unroll
        for (int k = 0; k < 6; ++k) xmax = fmaxf(xmax, mg[k] * 0.5f);
        float e[6], den = 0.f;
#pragma unroll
        for (int k = 0; k < 6; ++k) { e[k] = __expf(mg[k] * 0.5f - xmax); den += e[k]; }
        float inv = 1.0f / den;
        float loss = 0.f;
#pragma unroll
        for (int k = 0; k < 6; ++k) {
            float w2 = 0.8f * tg[k] / maxp;             // ALPHA * tgt / max_preds
            float ls = (1.0f - w2) * ce + w2 * kd[k];
            loss += e[k] * inv * ls;
        }
        acc += (double)loss;
    }
    for (int off = 16; off; off >>= 1) acc += __shfl_xor(acc, off, 32);
    if (lane == 0) ssum[wv] = acc;
    __syncthreads();
    if (tid == 0) {
        double t = 0.0;
        for (int w = 0; w < 8; ++w) t += ssum[w];
        out[0] = (float)(t / (double)BB);
    }
}

extern "C" void kernel_launch(void* const* d_in, const int* in_sizes, int n_in,
                              void* d_out, int out_size, void* d_ws, size_t ws_size,
                              hipStream_t stream) {
    const float* t0 = (const float*)d_in[0];
    const float* t1 = (const float*)d_in[1];
    const float* t2 = (const float*)d_in[2];
    const float* t3 = (const float*)d_in[3];
    const float* t4 = (const float*)d_in[4];
    const float* s  = (const float*)d_in[5];
    const int*   tg = (const int*)d_in[6];
    float* ws  = (float*)d_ws;   // 8192*20 f32 per-row stats + 512 f32 block maxes (~657 KB)
    float* out = (float*)d_out;

    const size_t lds = (size_t)3 * ROWS * RS * sizeof(float);  // 192,768 B dynamic LDS
    kd_row_kernel<<<dim3(BB / ROWS), dim3(BLK1), lds, stream>>>(t0, t1, t2, t3, t4, s, tg, ws);
    kd_final_kernel<<<dim3(1), dim3(256), 0, stream>>>(ws, out);
}